// LocalGrouper_9242769621759
// MI455X (gfx1250) — compile-verified
//
#include <hip/hip_runtime.h>
#include <hip/hip_bf16.h>

#define BATCH 8
#define NPTS  4096
#define NCH   128
#define NGRP  1024
#define KNN   32
#define GCH   (NCH + 3)        // 131
#define OUTCH (2 * NCH + 3)    // 259
#define EPSV  1e-5f
#define CHUNK 512

typedef __attribute__((ext_vector_type(2))) float v2f;
typedef __attribute__((ext_vector_type(8))) float v8f;

// wave32 argmin reduction on (value, index, aux); ties -> lowest index
__device__ __forceinline__ void argminRed(float& v, int& i, int& aux) {
#pragma unroll
  for (int off = 16; off > 0; off >>= 1) {
    float ov = __shfl_xor(v, off, 32);
    int   oi = __shfl_xor(i, off, 32);
    int   oa = __shfl_xor(aux, off, 32);
    if (ov < v || (ov == v && oi < i)) { v = ov; i = oi; aux = oa; }
  }
}

// wave32 argmax reduction on (value, index); ties -> lowest index (jnp.argmax)
__device__ __forceinline__ void argmaxRed(float& v, int& i) {
#pragma unroll
  for (int off = 16; off > 0; off >>= 1) {
    float ov = __shfl_xor(v, off, 32);
    int   oi = __shfl_xor(i, off, 32);
    if (ov > v || (ov == v && oi < i)) { v = ov; i = oi; }
  }
}

// ---------------------------------------------------------------------------
// Kernel 1: furthest point sampling. 1 block per batch, 1024 threads,
// 4 points/thread held entirely in registers (coords + running min-dist).
// Also emits new_xyz directly into d_out.
// ---------------------------------------------------------------------------
__global__ __launch_bounds__(1024)
void fps_kernel(const float* __restrict__ xyz, int* __restrict__ fps,
                float* __restrict__ newxyz) {
  const int b = blockIdx.x;
  const int tid = threadIdx.x;
  const int lane = tid & 31, wave = tid >> 5;
  __shared__ float rv[32];
  __shared__ int   ri[32];
  __shared__ int   s_last;
  const float* bx = xyz + (size_t)b * NPTS * 3;

  float px[4], py[4], pz[4], md[4];
#pragma unroll
  for (int j = 0; j < 4; ++j) {
    const int i = tid + j * 1024;
    px[j] = bx[i * 3 + 0];
    py[j] = bx[i * 3 + 1];
    pz[j] = bx[i * 3 + 2];
    md[j] = __builtin_inff();
  }
  if (tid == 0) s_last = 0;
  __syncthreads();

  for (int s = 0; s < NGRP; ++s) {
    const int last = s_last;
    const float qx = bx[last * 3 + 0];
    const float qy = bx[last * 3 + 1];
    const float qz = bx[last * 3 + 2];
    if (tid == 0) {
      fps[b * NGRP + s] = last;
      newxyz[(size_t)(b * NGRP + s) * 3 + 0] = qx;
      newxyz[(size_t)(b * NGRP + s) * 3 + 1] = qy;
      newxyz[(size_t)(b * NGRP + s) * 3 + 2] = qz;
    }
    float bv = -1.0f;
    int   bi = 0x7fffffff;
#pragma unroll
    for (int j = 0; j < 4; ++j) {
      const float dx = px[j] - qx, dy = py[j] - qy, dz = pz[j] - qz;
      const float d = dx * dx + dy * dy + dz * dz;
      md[j] = fminf(md[j], d);
      const int idx = tid + j * 1024;
      if (md[j] > bv || (md[j] == bv && idx < bi)) { bv = md[j]; bi = idx; }
    }
    argmaxRed(bv, bi);
    if (lane == 0) { rv[wave] = bv; ri[wave] = bi; }
    __syncthreads();
    if (wave == 0) {
      float v = rv[lane];
      int   i2 = ri[lane];
      argmaxRed(v, i2);
      if (lane == 0) s_last = i2;
    }
    __syncthreads();
  }
}

// ---------------------------------------------------------------------------
// Kernel 2: kNN via WMMA f32 16x16x4 distance tiles + per-wave top-32.
// Block = 512 threads (16 waves) handles 16 query rows of one batch.
// Per 512-point chunk: 32 WMMA tiles -> 16x512 distance tile in LDS,
// then each wave extracts sorted top-32 of its row and merges into its
// running top-32 (registers only).
// ---------------------------------------------------------------------------
__global__ __launch_bounds__(512)
void knn_kernel(const float* __restrict__ xyz, const int* __restrict__ fps,
                int* __restrict__ knn) {
  const int tiles = NGRP / 16;            // 64 row-tiles per batch
  const int b    = blockIdx.x / tiles;
  const int tile = blockIdx.x % tiles;
  const int tid  = threadIdx.x, lane = tid & 31, wave = tid >> 5;
  const float* bx = xyz + (size_t)b * NPTS * 3;

  __shared__ float q[16][4];
  __shared__ float qn[16];
  __shared__ float pn[CHUNK];
  __shared__ float dist[16][CHUNK];       // 32 KB

  if (tid < 16) {
    const int qi = fps[b * NGRP + tile * 16 + tid];
    const float x = bx[qi * 3 + 0], y = bx[qi * 3 + 1], z = bx[qi * 3 + 2];
    q[tid][0] = x; q[tid][1] = y; q[tid][2] = z; q[tid][3] = 0.0f;
    qn[tid] = x * x + y * y + z * z;
  }
  __syncthreads();

  // A-matrix (16x4, f32): lane%16 = M row; VGPR0 = K0|K2, VGPR1 = K1|K3
  const int  mi = lane & 15;
  const bool lo = lane < 16;
  v2f a;
  a.x = q[mi][lo ? 0 : 2];
  a.y = q[mi][lo ? 1 : 3];

  float rvv = __builtin_inff();
  int   rii = 0x7fffffff;

  for (int cb = 0; cb < NPTS; cb += CHUNK) {
    {  // point norms for this chunk
      const float* p = bx + (size_t)(cb + tid) * 3;
      pn[tid] = p[0] * p[0] + p[1] * p[1] + p[2] * p[2];
    }
    __syncthreads();

    // --- distance tiles via WMMA: 2 tiles per wave ---
#pragma unroll
    for (int tt = 0; tt < 2; ++tt) {
      const int t    = wave * 2 + tt;
      const int nloc = t * 16 + mi;
      const float* p = bx + (size_t)(cb + nloc) * 3;
      // B-matrix (4x16, f32): lane%16 = N col; VGPR0 = K0|K2, VGPR1 = K1|K3
      v2f bb;
      bb.x = p[lo ? 0 : 2];
      bb.y = lo ? p[1] : 0.0f;
      v8f c = {};
      v8f d = __builtin_amdgcn_wmma_f32_16x16x4_f32(
          false, a, false, bb, (short)0, c, false, false);
      const float pno = pn[nloc];
#pragma unroll
      for (int r = 0; r < 8; ++r) {
        const int m = r + (lo ? 0 : 8);
        dist[m][nloc] = qn[m] + pno - 2.0f * d[r];
      }
    }
    __syncthreads();

    // --- per-wave top-32 extraction of its row (row index == wave) ---
    float chv = __builtin_inff();
    int   chi = 0x7fffffff;
    for (int it = 0; it < 32; ++it) {
      float bv = __builtin_inff();
      int   bn = 0x7fffffff;
#pragma unroll
      for (int j = 0; j < CHUNK / 32; ++j) {
        const int n = lane + j * 32;
        const float v = dist[wave][n];
        if (v < bv) { bv = v; bn = n; }
      }
      int gi = cb + bn;
      argminRed(bv, gi, bn);
      if (lane == 0) dist[wave][bn] = __builtin_inff();
      if (lane == it) { chv = bv; chi = gi; }
      asm volatile("s_wait_dscnt 0" ::: "memory");  // intra-wave LDS fence
    }

    // --- merge chunk top-32 into running top-32 (registers only) ---
    float s0v = rvv;  int s0i = rii;
    float s1v = chv;  int s1i = chi;
    float nrv = __builtin_inff();
    int   nri = 0x7fffffff;
    for (int it = 0; it < 32; ++it) {
      const bool p1 = (s1v < s0v) || (s1v == s0v && s1i < s0i);
      float bv = p1 ? s1v : s0v;
      int   bi = p1 ? s1i : s0i;
      int   aux = (lane << 1) | (p1 ? 1 : 0);
      argminRed(bv, bi, aux);
      if ((aux >> 1) == lane) {
        if (aux & 1) s1v = __builtin_inff(); else s0v = __builtin_inff();
      }
      if (lane == it) { nrv = bv; nri = bi; }
    }
    rvv = nrv; rii = nri;
    __syncthreads();
  }

  // lane holds the lane-th nearest (ascending distance, lowest-idx ties)
  knn[((size_t)b * NGRP + tile * 16 + wave) * KNN + lane] = rii;
}

// ---------------------------------------------------------------------------
// Kernel 3a: per-group channel means + per-batch sum of squared centered
// values (global mean of centered is exactly 0, so std = sqrt(ssq/(M-1))).
// ---------------------------------------------------------------------------
__global__ __launch_bounds__(128)
void stats_kernel(const float* __restrict__ xyz, const float* __restrict__ points,
                  const int* __restrict__ knn, float* __restrict__ mean,
                  float* __restrict__ ssq) {
  const int bs = blockIdx.x;
  const int b  = bs / NGRP;
  const int tid = threadIdx.x, lane = tid & 31, wave = tid >> 5;
  __shared__ int   nk[KNN];
  __shared__ float red[4];
  if (tid < KNN) nk[tid] = knn[(size_t)bs * KNN + tid];
  __syncthreads();

  float lss = 0.0f;
  for (int ch = tid; ch < GCH; ch += 128) {
    float vals[KNN];
#pragma unroll
    for (int k = 0; k < KNN; ++k) {
      const int n = nk[k];
      vals[k] = (ch < NCH) ? points[((size_t)b * NPTS + n) * NCH + ch]
                           : xyz[((size_t)b * NPTS + n) * 3 + (ch - NCH)];
    }
    float sum = 0.0f;
#pragma unroll
    for (int k = 0; k < KNN; ++k) sum += vals[k];
    const float mu = sum * (1.0f / KNN);
    mean[(size_t)bs * GCH + ch] = mu;
#pragma unroll
    for (int k = 0; k < KNN; ++k) { const float d = vals[k] - mu; lss += d * d; }
  }
#pragma unroll
  for (int off = 16; off > 0; off >>= 1) lss += __shfl_xor(lss, off, 32);
  if (lane == 0) red[wave] = lss;
  __syncthreads();
  if (tid == 0) atomicAdd(&ssq[b], red[0] + red[1] + red[2] + red[3]);
}

// ---------------------------------------------------------------------------
// Kernel 3b: assemble output [B,S,K,259] = affine((g-mean)/(std+eps)) ++ anchor
// ---------------------------------------------------------------------------
__global__ __launch_bounds__(256)
void out_kernel(const float* __restrict__ xyz, const float* __restrict__ points,
                const float* __restrict__ alpha, const float* __restrict__ beta,
                const int* __restrict__ fps, const int* __restrict__ knn,
                const float* __restrict__ mean, const float* __restrict__ ssq,
                float* __restrict__ outp) {
  const int bs = blockIdx.x;
  const int b  = bs / NGRP;
  const int tid = threadIdx.x;
  const float mden = 1.0f / (float)((size_t)NGRP * KNN * GCH - 1);  // ddof=1
  const float stdv = sqrtf(ssq[b] * mden);
  const float inv  = 1.0f / (stdv + EPSV);
  const int anchor = fps[bs];
  float* orow = outp + (size_t)bs * KNN * OUTCH;

  for (int e = tid; e < KNN * OUTCH; e += 256) {
    const int k  = e / OUTCH;
    const int ch = e - k * OUTCH;
    float o;
    if (ch < GCH) {
      const int n = knn[(size_t)bs * KNN + k];
      const float g = (ch < NCH) ? points[((size_t)b * NPTS + n) * NCH + ch]
                                 : xyz[((size_t)b * NPTS + n) * 3 + (ch - NCH)];
      o = alpha[ch] * (g - mean[(size_t)bs * GCH + ch]) * inv + beta[ch];
    } else {
      o = points[((size_t)b * NPTS + anchor) * NCH + (ch - GCH)];
    }
    orow[e] = o;
  }
}

// ---------------------------------------------------------------------------
extern "C" void kernel_launch(void* const* d_in, const int* in_sizes, int n_in,
                              void* d_out, int out_size, void* d_ws, size_t ws_size,
                              hipStream_t stream) {
  (void)in_sizes; (void)n_in; (void)out_size; (void)ws_size;
  const float* xyz    = (const float*)d_in[0];
  const float* points = (const float*)d_in[1];
  const float* alpha  = (const float*)d_in[2];
  const float* beta   = (const float*)d_in[3];

  float* newxyz = (float*)d_out;                               // [B,S,3]
  float* outp   = (float*)d_out + (size_t)BATCH * NGRP * 3;    // [B,S,K,259]

  // workspace layout: fps_idx | knn_idx | mean | ssq   (~5.4 MB)
  int*   fps  = (int*)d_ws;                                    // B*S
  int*   knn  = fps + (size_t)BATCH * NGRP;                    // B*S*K
  float* mean = (float*)(knn + (size_t)BATCH * NGRP * KNN);    // B*S*131
  float* ssq  = mean + (size_t)BATCH * NGRP * GCH;             // B

  fps_kernel<<<dim3(BATCH), dim3(1024), 0, stream>>>(xyz, fps, newxyz);
  knn_kernel<<<dim3(BATCH * (NGRP / 16)), dim3(512), 0, stream>>>(xyz, fps, knn);
  hipMemsetAsync(ssq, 0, BATCH * sizeof(float), stream);
  stats_kernel<<<dim3(BATCH * NGRP), dim3(128), 0, stream>>>(xyz, points, knn, mean, ssq);
  out_kernel<<<dim3(BATCH * NGRP), dim3(256), 0, stream>>>(xyz, points, alpha, beta,
                                                           fps, knn, mean, ssq, outp);
}